// CenterLoss_7009386627592
// MI455X (gfx1250) — compile-verified
//
#include <hip/hip_runtime.h>

#ifndef __has_builtin
#define __has_builtin(x) 0
#endif

#if __has_builtin(__builtin_amdgcn_global_load_async_to_lds_b128)
#define ASYNC_LDS 1
#else
#define ASYNC_LDS 0
#endif

#if __has_builtin(__builtin_amdgcn_wmma_f32_16x16x4_f32)
#define WMMA_SUM 1
#else
#define WMMA_SUM 0
#endif

#define FEAT_D 512
#define TPB    256          // threads per class-block (8 waves on wave32)
#define SROWS  8            // gathered rows per stage
#define CHUNKS 4            // (SROWS*FEAT_D*4 bytes) / (TPB*16 bytes) = 4 b128 per thread

typedef int   v4i_t __attribute__((ext_vector_type(4)));
typedef float v2f_t __attribute__((ext_vector_type(2)));
typedef float v8f_t __attribute__((ext_vector_type(8)));

// ---- async global->LDS helpers (gfx1250 path, guarded) ---------------------
__device__ __forceinline__ void stage16(const float* g, float* l) {
#if ASYNC_LDS
    __builtin_amdgcn_global_load_async_to_lds_b128(
        (__attribute__((address_space(1))) v4i_t*)g,
        (__attribute__((address_space(3))) v4i_t*)l,
        0, 0);
#else
    *reinterpret_cast<float4*>(l) = *reinterpret_cast<const float4*>(g);
#endif
}

template <int N>
__device__ __forceinline__ void wait_async() {
#if ASYNC_LDS
#if __has_builtin(__builtin_amdgcn_s_wait_asynccnt)
    __builtin_amdgcn_s_wait_asynccnt(N);
#else
    asm volatile("s_wait_asynccnt %0" ::"i"(N) : "memory");
#endif
#endif
}

// ---- phase 1: zero the histogram ------------------------------------------
__global__ void cl_zero_kernel(int* __restrict__ counts, int C) {
    int i = blockIdx.x * blockDim.x + threadIdx.x;
    if (i < C) counts[i] = 0;
}

// ---- phase 2: per-class counts (integer atomics: deterministic) -----------
__global__ void cl_count_kernel(const int* __restrict__ labels,
                                int* __restrict__ counts, int B) {
    int i = blockIdx.x * blockDim.x + threadIdx.x;
    if (i < B) atomicAdd(&counts[labels[i]], 1);
}

// ---- phase 3: exclusive scan over C<=1024 counts (single block) -----------
__global__ void cl_scan_kernel(const int* __restrict__ counts,
                               int* __restrict__ offsets,
                               int* __restrict__ cursor, int C) {
    __shared__ int sh[1024];
    int t = threadIdx.x;
    int v = (t < C) ? counts[t] : 0;
    sh[t] = v;
    __syncthreads();
    for (int d = 1; d < 1024; d <<= 1) {
        int add = (t >= d) ? sh[t - d] : 0;
        __syncthreads();
        sh[t] += add;
        __syncthreads();
    }
    if (t < C) {
        int excl = sh[t] - v;   // exclusive prefix
        offsets[t] = excl;
        cursor[t]  = excl;
    }
}

// ---- phase 4: counting-sort permutation (group sample ids by class) -------
__global__ void cl_permute_kernel(const int* __restrict__ labels,
                                  int* __restrict__ cursor,
                                  int* __restrict__ perm, int B) {
    int i = blockIdx.x * blockDim.x + threadIdx.x;
    if (i < B) {
        int p = atomicAdd(&cursor[labels[i]], 1);
        perm[p] = i;
    }
}

// ---- phase 5: one workgroup per class; async-staged gather of x rows ------
// Column sums Sigma_j x[j][d] are computed on the matrix pipe via
// V_WMMA_F32_16X16X4_F32 with an all-ones A matrix (D[m,n] = Sigma_k B[k,n],
// accumulated into C). VALU keeps only the loss (residual^2) math.
__global__ __launch_bounds__(TPB) void cl_class_kernel(
    const float* __restrict__ x, const float* __restrict__ centers,
    const int* __restrict__ counts, const int* __restrict__ offsets,
    const int* __restrict__ perm, float* __restrict__ new_centers,
    float* __restrict__ loss_pc) {
    __shared__ __align__(16) float buf[2][SROWS * FEAT_D]; // 32 KB double buffer
    __shared__ float red[TPB];

    const int c    = blockIdx.x;
    const int t    = threadIdx.x;
    const int lane = t & 31;
    const int wave = t >> 5;
    const int cnt  = counts[c];
    const int off  = offsets[c];

    const float cc0 = centers[c * FEAT_D + 2 * t];
    const float cc1 = centers[c * FEAT_D + 2 * t + 1];

    if (cnt == 0) {                       // uniform branch per block
        new_centers[c * FEAT_D + 2 * t]     = cc0;
        new_centers[c * FEAT_D + 2 * t + 1] = cc1;
        if (t == 0) loss_pc[c] = 0.0f;
        return;
    }

    auto prefetch = [&](int s) {
        // SROWS rows * 128 b128-chunks/row, covered by TPB threads * CHUNKS
        #pragma unroll
        for (int q = 0; q < CHUNKS; ++q) {
            int lin = q * TPB + t;        // 0 .. SROWS*128-1
            int j   = lin >> 7;           // staged row slot
            int col = lin & 127;          // 16B chunk within the row
            int gi  = s * SROWS + j;
            if (gi >= cnt) gi = cnt - 1;  // clamp: uniform async issue count
            int row = perm[off + gi];
            const float* g = x + (size_t)row * FEAT_D + col * 4;
            float* l = &buf[s & 1][j * FEAT_D + col * 4];
            stage16(g, l);
        }
    };

    const int nst = (cnt + SROWS - 1) / SROWS;
    float lsum = 0.0f;

#if WMMA_SUM
    v8f_t wacc[4];                        // 4 groups of 16 features per wave
    #pragma unroll
    for (int g = 0; g < 4; ++g) wacc[g] = (v8f_t)0.0f;
    const int nh = lane >> 4;             // K-half of this lane in B operand
    const int nl = lane & 15;             // N (feature) index of this lane
#else
    float acc0 = 0.0f, acc1 = 0.0f;
#endif

    prefetch(0);
    for (int s = 0; s < nst; ++s) {
        if (s + 1 < nst) { prefetch(s + 1); wait_async<CHUNKS>(); }
        else             { wait_async<0>(); }
        __syncthreads();                  // stage (s&1) complete for all waves

        float* bm = buf[s & 1];
        const int rows = min(SROWS, cnt - s * SROWS);
#if WMMA_SUM
        if (rows < SROWS) {
            // zero junk duplicate rows so fixed-K WMMA sums stay exact
            for (int idx = t; idx < (SROWS - rows) * FEAT_D; idx += TPB)
                bm[rows * FEAT_D + idx] = 0.0f;
            __syncthreads();
        }
#endif
        const float* b = bm;

        // loss: each thread owns features 2t, 2t+1
        for (int j = 0; j < rows; ++j) {
            float v0 = b[j * FEAT_D + 2 * t];
            float v1 = b[j * FEAT_D + 2 * t + 1];
            float d0 = v0 - cc0, d1 = v1 - cc1;
            lsum = fmaf(d0, d0, lsum);
            lsum = fmaf(d1, d1, lsum);
#if !WMMA_SUM
            acc0 += v0;
            acc1 += v1;
#endif
        }

#if WMMA_SUM
        // column sums on the matrix pipe: 2 K-tiles (rows 0-3, 4-7) x 4 groups
        #pragma unroll
        for (int g = 0; g < 4; ++g) {
            const int f = wave * 64 + g * 16 + nl;
            #pragma unroll
            for (int kt = 0; kt < 2; ++kt) {
                const int k0 = kt * 4 + nh * 2;   // lane's K pair within tile
                v2f_t bv;
                bv.x = b[(k0 + 0) * FEAT_D + f];
                bv.y = b[(k0 + 1) * FEAT_D + f];
                v2f_t av;
                av.x = 1.0f;
                av.y = 1.0f;
                wacc[g] = __builtin_amdgcn_wmma_f32_16x16x4_f32(
                    false, av, false, bv, (short)0, wacc[g], false, false);
            }
        }
#endif
        __syncthreads();                  // protect buffer before re-staging
    }

    // new_center = c - 0.5 * (cnt*c - sum(x)) / cnt
    const float fc  = (float)cnt;
    const float inv = 1.0f / fc;
#if WMMA_SUM
    #pragma unroll
    for (int g = 0; g < 4; ++g) {
        if (lane < 16) {
            const int f = wave * 64 + g * 16 + lane;
            float s  = wacc[g][0];        // M=0 row, N=lane (replicated over M)
            float cc = centers[c * FEAT_D + f];
            new_centers[c * FEAT_D + f] = cc - 0.5f * (fc * cc - s) * inv;
        }
    }
#else
    new_centers[c * FEAT_D + 2 * t]     = cc0 - 0.5f * (fc * cc0 - acc0) * inv;
    new_centers[c * FEAT_D + 2 * t + 1] = cc1 - 0.5f * (fc * cc1 - acc1) * inv;
#endif

    // block-reduce the squared-distance partials (fixed order: deterministic)
    red[t] = lsum;
    __syncthreads();
    for (int st = TPB / 2; st > 0; st >>= 1) {
        if (t < st) red[t] += red[t + st];
        __syncthreads();
    }
    if (t == 0) loss_pc[c] = red[0];
}

// ---- phase 6: fixed-order reduction of per-class loss ---------------------
__global__ void cl_loss_final_kernel(const float* __restrict__ loss_pc,
                                     float* __restrict__ out, int C,
                                     float scale) {
    __shared__ float sh[1024];
    int t = threadIdx.x;
    sh[t] = (t < C) ? loss_pc[t] : 0.0f;
    __syncthreads();
    for (int st = 512; st > 0; st >>= 1) {
        if (t < st) sh[t] += sh[t + st];
        __syncthreads();
    }
    if (t == 0) out[0] = sh[0] * scale;
}

extern "C" void kernel_launch(void* const* d_in, const int* in_sizes, int n_in,
                              void* d_out, int out_size, void* d_ws, size_t ws_size,
                              hipStream_t stream) {
    const float* x       = (const float*)d_in[0];
    const int*   labels  = (const int*)d_in[1];
    const float* centers = (const float*)d_in[2];

    const int B = in_sizes[1];             // 65536
    const int D = in_sizes[0] / B;         // 512 (== FEAT_D)
    const int C = in_sizes[2] / D;         // 1000

    float* out = (float*)d_out;            // out[0]=loss, out+1 = new_centers

    int*   ws      = (int*)d_ws;
    int*   counts  = ws;
    int*   offsets = ws + C;
    int*   cursor  = ws + 2 * C;
    int*   perm    = ws + 3 * C;
    float* loss_pc = (float*)(ws + 3 * C + B);

    cl_zero_kernel   <<<(C + 255) / 256, 256, 0, stream>>>(counts, C);
    cl_count_kernel  <<<(B + 255) / 256, 256, 0, stream>>>(labels, counts, B);
    cl_scan_kernel   <<<1, 1024, 0, stream>>>(counts, offsets, cursor, C);
    cl_permute_kernel<<<(B + 255) / 256, 256, 0, stream>>>(labels, cursor, perm, B);
    cl_class_kernel  <<<C, TPB, 0, stream>>>(x, centers, counts, offsets, perm,
                                             out + 1, loss_pc);
    cl_loss_final_kernel<<<1, 1024, 0, stream>>>(loss_pc, out, C, 0.5f / (float)B);
}